// BlindPnPNeuralSolver_80333068304777
// MI455X (gfx1250) — compile-verified
//
#include <hip/hip_runtime.h>
#include <math.h>

#define MPTS 4096
#define NPTS 4096
#define FDIM 128
#define NITER 200
#define MU_INV 10.0f   // 1/0.1
#define CHUNKS 64      // row chunks for K^T u partial sums (4096/64 = 64 rows each)

typedef float v2f __attribute__((ext_vector_type(2)));
typedef float v8f __attribute__((ext_vector_type(8)));

// ---------------------------------------------------------------------------
// WMMA fp32 16x16x4 tile: 16x16 tile of A (MxK, row-major) times B^T where
// B is (NxK, row-major), K stepped by 4.
// fp32 A layout (ISA 7.12.2): lanes 0-15 hold M=lane with K0/K1 in v0/v1;
// lanes 16-31 hold K2/K3. B (4x16) mirrors this with N striped across lanes.
// ---------------------------------------------------------------------------
__device__ __forceinline__ v8f wmma_abT_tile(const float* __restrict__ A,
                                             const float* __restrict__ B,
                                             int lda, int ldb, int K,
                                             unsigned lane) {
  v8f acc = {0.f, 0.f, 0.f, 0.f, 0.f, 0.f, 0.f, 0.f};
  const int m  = lane & 15;            // row (A) / col (B) index within tile
  const int ko = (lane >> 4) << 1;     // 0 or 2: which K-pair this half-wave holds
  const float* a = A + (size_t)m * lda + ko;
  const float* b = B + (size_t)m * ldb + ko;
  for (int k = 0; k < K; k += 4) {
    v2f av, bv;
    av.x = a[k];     av.y = a[k + 1];
    bv.x = b[k];     bv.y = b[k + 1];
    acc = __builtin_amdgcn_wmma_f32_16x16x4_f32(false, av, false, bv,
                                                (short)0, acc, false, false);
  }
  return acc;
}

__device__ __forceinline__ float sigmoid_f(float x) {
  return 1.0f / (1.0f + expf(-x));
}

// ---------------------------------------------------------------------------
// Input prep: l2-normalized surface normals + bearing vectors
// ---------------------------------------------------------------------------
__global__ void prep2d_kernel(const float* __restrict__ sn2d,
                              const float* __restrict__ pix2d,
                              const float* __restrict__ intr,
                              float* __restrict__ X) {
  int row = blockIdx.x * blockDim.x + threadIdx.x;
  float s0 = sn2d[row * 3 + 0], s1 = sn2d[row * 3 + 1], s2 = sn2d[row * 3 + 2];
  float ninv = 1.0f / fmaxf(sqrtf(s0 * s0 + s1 * s1 + s2 * s2), 1e-12f);

  // analytic 3x3 inverse of intrinsics (adjugate / det)
  float a = intr[0], b = intr[1], c = intr[2];
  float d = intr[3], e = intr[4], f = intr[5];
  float g = intr[6], h = intr[7], i = intr[8];
  float det = a * (e * i - f * h) - b * (d * i - f * g) + c * (d * h - e * g);
  float id = 1.0f / det;
  float i00 = (e * i - f * h) * id, i01 = (c * h - b * i) * id, i02 = (b * f - c * e) * id;
  float i10 = (f * g - d * i) * id, i11 = (a * i - c * g) * id, i12 = (c * d - a * f) * id;
  float i20 = (d * h - e * g) * id, i21 = (b * g - a * h) * id, i22 = (a * e - b * d) * id;

  float px = pix2d[row * 2 + 0], py = pix2d[row * 2 + 1];
  float b0 = i00 * px + i01 * py + i02;
  float b1 = i10 * px + i11 * py + i12;
  float b2 = i20 * px + i21 * py + i22;
  // index reorder [1,0,2]
  float v0 = b1, v1 = b0, v2 = b2;
  float binv = 1.0f / fmaxf(sqrtf(v0 * v0 + v1 * v1 + v2 * v2), 1e-12f);

  float* x = X + (size_t)row * 6;
  x[0] = s0 * ninv; x[1] = s1 * ninv; x[2] = s2 * ninv;
  x[3] = v0 * binv; x[4] = v1 * binv; x[5] = v2 * binv;
}

__global__ void prep3d_kernel(const float* __restrict__ sn3d,
                              const float* __restrict__ pts3d,
                              float* __restrict__ X) {
  int row = blockIdx.x * blockDim.x + threadIdx.x;
  float s0 = sn3d[row * 3 + 0], s1 = sn3d[row * 3 + 1], s2 = sn3d[row * 3 + 2];
  float ninv = 1.0f / fmaxf(sqrtf(s0 * s0 + s1 * s1 + s2 * s2), 1e-12f);
  float p0 = pts3d[row * 3 + 0], p1 = pts3d[row * 3 + 1], p2 = pts3d[row * 3 + 2];
  float pinv = 1.0f / fmaxf(sqrtf(p0 * p0 + p1 * p1 + p2 * p2), 1e-12f);
  float* x = X + (size_t)row * 6;
  x[0] = s0 * ninv; x[1] = s1 * ninv; x[2] = s2 * ninv;
  x[3] = p0 * pinv; x[4] = p1 * pinv; x[5] = p2 * pinv;
}

// ---------------------------------------------------------------------------
// MLP layer 1 (K=6, too small for WMMA): thread per (row, out)
// ---------------------------------------------------------------------------
__global__ void layer1_kernel(const float* __restrict__ X,
                              const float* __restrict__ W,
                              const float* __restrict__ bias,
                              float* __restrict__ H) {
  int idx = blockIdx.x * blockDim.x + threadIdx.x;   // 4096*64
  int row = idx >> 6;
  int o = idx & 63;
  const float* x = X + (size_t)row * 6;
  const float* w = W + (size_t)o * 6;
  float s = bias[o];
#pragma unroll
  for (int k = 0; k < 6; ++k) s = fmaf(x[k], w[k], s);
  H[(size_t)row * 64 + o] = sigmoid_f(s);
}

// ---------------------------------------------------------------------------
// WMMA GEMM: Out = sigmoid(A @ B^T + bias); one 16x16 tile per wave.
// ---------------------------------------------------------------------------
__global__ void gemm_abT_bias_sigmoid(const float* __restrict__ A, int lda,
                                      const float* __restrict__ B, int ldb,
                                      const float* __restrict__ bias,
                                      float* __restrict__ Out, int ldo,
                                      int K, int ntn) {
  unsigned lane = threadIdx.x & 31u;
  int wave = (int)((blockIdx.x * blockDim.x + threadIdx.x) >> 5);
  int tm = wave / ntn, tn = wave % ntn;
  v8f acc = wmma_abT_tile(A + (size_t)(tm * 16) * lda,
                          B + (size_t)(tn * 16) * ldb, lda, ldb, K, lane);
  int col = tn * 16 + (int)(lane & 15);
  int rbase = tm * 16 + (int)((lane >> 4) << 3);   // C/D: vgpr j -> M=j (+8 for hi lanes)
  float bv = bias[col];
#pragma unroll
  for (int j = 0; j < 8; ++j) {
    Out[(size_t)(rbase + j) * ldo + col] = sigmoid_f(acc[j] + bv);
  }
}

// ---------------------------------------------------------------------------
// Row-wise l2 normalize (128-wide rows); one wave per row (wave32 shuffles).
// Also emits post-normalization squared norm (for pairwise_l2 x2/y2 terms).
// ---------------------------------------------------------------------------
__global__ void l2norm_rows_kernel(float* __restrict__ F, float* __restrict__ nsq) {
  unsigned lane = threadIdx.x & 31u;
  int row = (int)((blockIdx.x * blockDim.x + threadIdx.x) >> 5);
  float* f = F + (size_t)row * FDIM;
  float s = 0.0f;
#pragma unroll
  for (int t = 0; t < 4; ++t) { float x = f[lane + 32 * t]; s = fmaf(x, x, s); }
#pragma unroll
  for (int off = 16; off; off >>= 1) s += __shfl_xor(s, off, 32);
  float inv = 1.0f / fmaxf(sqrtf(s), 1e-12f);
#pragma unroll
  for (int t = 0; t < 4; ++t) f[lane + 32 * t] *= inv;
  if (lane == 0) nsq[row] = s * inv * inv;
}

// ---------------------------------------------------------------------------
// Pairwise L2 + exp(-M/mu), fused: K[i][j] = exp(-sqrt(max(x2+y2-2*dot,eps))/mu)
// WMMA f32 for the 4096x4096x128 dot products; one 16x16 tile per wave.
// ---------------------------------------------------------------------------
__global__ void pairwise_expK_kernel(const float* __restrict__ F2,
                                     const float* __restrict__ F3,
                                     const float* __restrict__ n2sq,
                                     const float* __restrict__ n3sq,
                                     float* __restrict__ Kout) {
  unsigned lane = threadIdx.x & 31u;
  int wave = (int)((blockIdx.x * blockDim.x + threadIdx.x) >> 5);
  int tm = wave >> 8, tn = wave & 255;   // 256x256 tiles
  v8f acc = wmma_abT_tile(F2 + (size_t)(tm * 16) * FDIM,
                          F3 + (size_t)(tn * 16) * FDIM, FDIM, FDIM, FDIM, lane);
  int col = tn * 16 + (int)(lane & 15);
  int rbase = tm * 16 + (int)((lane >> 4) << 3);
  float y2 = n3sq[col];
#pragma unroll
  for (int j = 0; j < 8; ++j) {
    int row = rbase + j;
    float d2 = fmaxf(n2sq[row] + y2 - 2.0f * acc[j], 1e-12f);
    Kout[(size_t)row * NPTS + col] = expf(-sqrtf(d2) * MU_INV);
  }
}

// ---------------------------------------------------------------------------
// Sinkhorn: K (64MB fp32) lives in d_out, resident in the 192MB L2.
// All streaming kernels use float4 (global_load_b128) to minimize issue
// slots per byte on the bandwidth-bound path.
// ---------------------------------------------------------------------------
__global__ void init_u_kernel(float* __restrict__ u) {
  int i = blockIdx.x * blockDim.x + threadIdx.x;
  u[i] = 1.0f;
}

// phase A of v = c / (K^T u): partial[chunk][col] over 64-row chunks.
// Each thread owns 4 consecutive columns (float4); coalesced b128 reads.
__global__ void colsum_partial_kernel(const float4* __restrict__ Km4,
                                      const float* __restrict__ u,
                                      float4* __restrict__ partial4) {
  int c4 = blockIdx.x * blockDim.x + threadIdx.x;    // 0..1023 float4 columns
  int ch = blockIdx.y;                               // 0..CHUNKS-1
  int i0 = ch * (NPTS / CHUNKS);
  float4 s = {0.f, 0.f, 0.f, 0.f};
  for (int i = i0; i < i0 + (NPTS / CHUNKS); ++i) {
    __builtin_prefetch(&Km4[(size_t)(i + 8) * (NPTS / 4) + c4], 0, 0);
    float4 k = Km4[(size_t)i * (NPTS / 4) + c4];
    float ui = u[i];
    s.x = fmaf(k.x, ui, s.x);
    s.y = fmaf(k.y, ui, s.y);
    s.z = fmaf(k.z, ui, s.z);
    s.w = fmaf(k.w, ui, s.w);
  }
  partial4[(size_t)ch * (NPTS / 4) + c4] = s;
}

__global__ void v_update_kernel(const float4* __restrict__ partial4,
                                float4* __restrict__ v4) {
  int c4 = blockIdx.x * blockDim.x + threadIdx.x;    // 0..1023
  float4 s = {0.f, 0.f, 0.f, 0.f};
#pragma unroll
  for (int ch = 0; ch < CHUNKS; ++ch) {
    float4 p = partial4[(size_t)ch * (NPTS / 4) + c4];
    s.x += p.x; s.y += p.y; s.z += p.z; s.w += p.w;
  }
  const float c = 1.0f / (float)NPTS;
  float4 r = {c / s.x, c / s.y, c / s.z, c / s.w};
  v4[c4] = r;
}

// u = r / (K v): one wave per row; lanes read float4s (512B/wave/issue),
// then wave32 shuffle reduction.
__global__ void u_update_kernel(const float4* __restrict__ Km4,
                                const float4* __restrict__ v4,
                                float* __restrict__ u) {
  unsigned lane = threadIdx.x & 31u;
  int row = (int)((blockIdx.x * blockDim.x + threadIdx.x) >> 5);
  const float4* krow = Km4 + (size_t)row * (NPTS / 4);
  float s = 0.0f;
#pragma unroll 4
  for (int t = 0; t < NPTS / 4 / 32; ++t) {
    int j = (int)lane + 32 * t;
    float4 k = krow[j];
    float4 vv = v4[j];
    s = fmaf(k.x, vv.x, s);
    s = fmaf(k.y, vv.y, s);
    s = fmaf(k.z, vv.z, s);
    s = fmaf(k.w, vv.w, s);
  }
#pragma unroll
  for (int off = 16; off; off >>= 1) s += __shfl_xor(s, off, 32);
  if (lane == 0) u[row] = (1.0f / (float)MPTS) / s;
}

// P = u[:,None] * K * v[None,:]  (in place over d_out, float4)
__global__ void scaleP_kernel(float4* __restrict__ Km4,
                              const float* __restrict__ u,
                              const float4* __restrict__ v4) {
  int c4 = blockIdx.x * blockDim.x + threadIdx.x;    // 0..1023
  int i = blockIdx.y;
  size_t idx = (size_t)i * (NPTS / 4) + c4;
  float ui = u[i];
  float4 k = Km4[idx];
  float4 vv = v4[c4];
  float4 p = {ui * k.x * vv.x, ui * k.y * vv.y, ui * k.z * vv.z, ui * k.w * vv.w};
  Km4[idx] = p;
}

// ---------------------------------------------------------------------------
extern "C" void kernel_launch(void* const* d_in, const int* in_sizes, int n_in,
                              void* d_out, int out_size, void* d_ws, size_t ws_size,
                              hipStream_t stream) {
  const float* sn2d  = (const float*)d_in[0];
  const float* sn3d  = (const float*)d_in[1];
  const float* pix2d = (const float*)d_in[2];
  const float* pts3d = (const float*)d_in[3];
  const float* intr  = (const float*)d_in[4];
  const float* W1i = (const float*)d_in[5];  const float* b1i = (const float*)d_in[6];
  const float* W2i = (const float*)d_in[7];  const float* b2i = (const float*)d_in[8];
  const float* W3i = (const float*)d_in[9];  const float* b3i = (const float*)d_in[10];
  const float* W1p = (const float*)d_in[11]; const float* b1p = (const float*)d_in[12];
  const float* W2p = (const float*)d_in[13]; const float* b2p = (const float*)d_in[14];
  const float* W3p = (const float*)d_in[15]; const float* b3p = (const float*)d_in[16];

  float*  Km  = (float*)d_out;    // 64MB K matrix lives in d_out, scaled to P in place
  float4* Km4 = (float4*)d_out;

  // workspace layout (~10 MB, fully rewritten each call)
  float* ws   = (float*)d_ws;
  float* x2d  = ws;                        // 4096*6
  float* x3d  = x2d + MPTS * 6;            // 4096*6
  float* hA   = x3d + NPTS * 6;            // 4096*128
  float* hB   = hA + MPTS * 128;           // 4096*128
  float* f2d  = hB + MPTS * 128;           // 4096*128
  float* f3d  = f2d + MPTS * 128;          // 4096*128
  float* n2sq = f3d + NPTS * 128;          // 4096
  float* n3sq = n2sq + MPTS;               // 4096
  float* u    = n3sq + NPTS;               // 4096
  float* v    = u + MPTS;                  // 4096
  float* part = v + NPTS;                  // CHUNKS*4096

  // ---- feature extraction ----
  prep2d_kernel<<<16, 256, 0, stream>>>(sn2d, pix2d, intr, x2d);
  prep3d_kernel<<<16, 256, 0, stream>>>(sn3d, pts3d, x3d);

  // image branch MLP
  layer1_kernel<<<1024, 256, 0, stream>>>(x2d, W1i, b1i, hA);
  gemm_abT_bias_sigmoid<<<256, 256, 0, stream>>>(hA, 64, W2i, 64, b2i, hB, 128, 64, 8);
  gemm_abT_bias_sigmoid<<<256, 256, 0, stream>>>(hB, 128, W3i, 128, b3i, f2d, 128, 128, 8);
  l2norm_rows_kernel<<<512, 256, 0, stream>>>(f2d, n2sq);

  // point branch MLP
  layer1_kernel<<<1024, 256, 0, stream>>>(x3d, W1p, b1p, hA);
  gemm_abT_bias_sigmoid<<<256, 256, 0, stream>>>(hA, 64, W2p, 64, b2p, hB, 128, 64, 8);
  gemm_abT_bias_sigmoid<<<256, 256, 0, stream>>>(hB, 128, W3p, 128, b3p, f3d, 128, 128, 8);
  l2norm_rows_kernel<<<512, 256, 0, stream>>>(f3d, n3sq);

  // ---- cost matrix + Gibbs kernel (WMMA GEMM, fused sqrt/exp) ----
  pairwise_expK_kernel<<<8192, 256, 0, stream>>>(f2d, f3d, n2sq, n3sq, Km);

  // ---- Sinkhorn (200 iterations; K streamed from L2) ----
  init_u_kernel<<<16, 256, 0, stream>>>(u);
  for (int it = 0; it < NITER; ++it) {
    colsum_partial_kernel<<<dim3(4, CHUNKS), 256, 0, stream>>>(Km4, u, (float4*)part);
    v_update_kernel<<<4, 256, 0, stream>>>((const float4*)part, (float4*)v);
    u_update_kernel<<<512, 256, 0, stream>>>(Km4, (const float4*)v, u);
  }
  // final v = c / (K^T u)
  colsum_partial_kernel<<<dim3(4, CHUNKS), 256, 0, stream>>>(Km4, u, (float4*)part);
  v_update_kernel<<<4, 256, 0, stream>>>((const float4*)part, (float4*)v);

  // P = diag(u) K diag(v), in place over d_out
  scaleP_kernel<<<dim3(4, NPTS), 256, 0, stream>>>(Km4, u, (const float4*)v);
}